// Transofrmer_2671469658530
// MI455X (gfx1250) — compile-verified
//
#include <hip/hip_runtime.h>

typedef float v2f __attribute__((ext_vector_type(2)));
typedef float v8f __attribute__((ext_vector_type(8)));

// ---------------------------------------------------------------------------
// Kernel 1: kq = x @ Wk^T + bk
//   x  : [32768, 512]  (B*R rows)
//   Wk : [256, 512]    (row = output channel e, so B-matrix[k][n] = Wk[n][k])
//   kq : [32768, 256]  == [B, R, H, Q] flattened
// One wave computes one 16x16 output tile with V_WMMA_F32_16X16X4_F32,
// looping K=512 in steps of 4. Wk tile (16x512 = 32KB) is L1/L2 resident.
// ---------------------------------------------------------------------------
__global__ __launch_bounds__(32)
void kq_gemm(const float* __restrict__ x,
             const float* __restrict__ Wk,
             const float* __restrict__ bk,
             float* __restrict__ kq)
{
    const int m0   = blockIdx.x * 16;
    const int n0   = blockIdx.y * 16;
    const int lane = threadIdx.x;   // 0..31 (wave32)
    const int half = lane >> 4;     // 0: K={0,1}, 1: K={2,3}
    const int l16  = lane & 15;

    // A fragment: lane holds row (m0+l16), consecutive K pair at k+2*half
    const float* arow = x  + (size_t)(m0 + l16) * 512 + 2 * half;
    // B fragment: lane holds column (n0+l16) of B = row of Wk, K pair at k+2*half
    const float* brow = Wk + (size_t)(n0 + l16) * 512 + 2 * half;

    v8f c = {0.f, 0.f, 0.f, 0.f, 0.f, 0.f, 0.f, 0.f};
#pragma unroll 8
    for (int k = 0; k < 512; k += 4) {
        v2f a = *(const v2f*)(arow + k);
        v2f b = *(const v2f*)(brow + k);
        // (neg_a, A, neg_b, B, c_mod, C, reuse_a, reuse_b)
        c = __builtin_amdgcn_wmma_f32_16x16x4_f32(false, a, false, b,
                                                  (short)0, c, false, false);
    }

    // D layout: VGPR p -> row m0 + p + 8*half, col n0 + l16
    const float bias = bk[n0 + l16];
    float* outp = kq + (size_t)(m0 + 8 * half) * 256 + n0 + l16;
#pragma unroll
    for (int p = 0; p < 8; ++p)
        outp[(size_t)p * 256] = c[p] + bias;
}

// ---------------------------------------------------------------------------
// Kernel 2 (fused): for one (b, r-tile, s-tile), compute QK^T/sqrt(32) for all
// 8 heads (8 WMMA accumulator sets with identical lane layout), then
// relu(x - thr) and softmax ACROSS HEADS entirely in registers, then store
// the 8 head tiles of the 1 GiB output. relation is never materialized.
// ---------------------------------------------------------------------------
__global__ __launch_bounds__(32)
void rel_softmax(const float* __restrict__ kq,
                 const float* __restrict__ thrp,
                 float* __restrict__ out)
{
    const int s0   = blockIdx.x * 16;
    const int r0   = blockIdx.y * 16;
    const int b    = blockIdx.z;
    const int lane = threadIdx.x;
    const int half = lane >> 4;
    const int l16  = lane & 15;

    const float thr   = thrp[0];
    const float scale = 0.17677669529663688f; // 1/sqrt(32)

    // A: rows r0..r0+15 of kq (lane = row), B: columns = rows s0..s0+15 of kq
    const float* Abase = kq + (size_t)(b * 1024 + r0 + l16) * 256 + 2 * half;
    const float* Bbase = kq + (size_t)(b * 1024 + s0 + l16) * 256 + 2 * half;

    v8f acc[8];
#pragma unroll
    for (int h = 0; h < 8; ++h) {
        v8f c = {0.f, 0.f, 0.f, 0.f, 0.f, 0.f, 0.f, 0.f};
        const float* ah = Abase + h * 32;
        const float* bh = Bbase + h * 32;
#pragma unroll
        for (int k = 0; k < 32; k += 4) {
            v2f a  = *(const v2f*)(ah + k);
            v2f bb = *(const v2f*)(bh + k);
            c = __builtin_amdgcn_wmma_f32_16x16x4_f32(false, a, false, bb,
                                                      (short)0, c, false, false);
        }
        acc[h] = c;
    }

    // out[b,h,r,s]: base at (b*8)*1024*1024 + (r0+8*half)*1024 + s0 + l16
    float* obase = out + ((size_t)b * 8 * 1024 + (size_t)(r0 + 8 * half)) * 1024
                       + s0 + l16;

#pragma unroll
    for (int p = 0; p < 8; ++p) {
        float v[8];
        float mx = 0.0f;                    // relu output >= 0, so max >= 0
#pragma unroll
        for (int h = 0; h < 8; ++h) {
            float t = acc[h][p] * scale - thr;
            t = t > 0.f ? t : 0.f;          // relu
            v[h] = t;
            mx = t > mx ? t : mx;
        }
        float sum = 0.f;
#pragma unroll
        for (int h = 0; h < 8; ++h) {
            v[h] = __expf(v[h] - mx);
            sum += v[h];
        }
        const float inv = 1.0f / sum;
#pragma unroll
        for (int h = 0; h < 8; ++h)
            obase[(size_t)h * 1024 * 1024 + (size_t)p * 1024] = v[h] * inv;
    }
}

// ---------------------------------------------------------------------------
extern "C" void kernel_launch(void* const* d_in, const int* in_sizes, int n_in,
                              void* d_out, int out_size, void* d_ws, size_t ws_size,
                              hipStream_t stream)
{
    const float* x   = (const float*)d_in[0];  // [32, 1024, 512]
    const float* Wk  = (const float*)d_in[1];  // [256, 512]
    const float* bk  = (const float*)d_in[2];  // [256]
    const float* thr = (const float*)d_in[3];  // [1]
    float* out = (float*)d_out;                // [32, 8, 1024, 1024]
    float* kq  = (float*)d_ws;                 // 32768*256 f32 = 33.5 MB scratch

    // Kernel 1: 2048 x 16 tiles of 16x16, one wave each
    hipLaunchKernelGGL(kq_gemm, dim3(32768 / 16, 256 / 16), dim3(32), 0, stream,
                       x, Wk, bk, kq);

    // Kernel 2: (s-tiles, r-tiles, batch), one wave each, 8 heads fused
    hipLaunchKernelGGL(rel_softmax, dim3(1024 / 16, 1024 / 16, 32), dim3(32), 0,
                       stream, kq, thr, out);
}